// SpectralNormError_15745350107848
// MI455X (gfx1250) — compile-verified
//
#include <hip/hip_runtime.h>
#include <hip/hip_bf16.h>
#include <math.h>

typedef __attribute__((ext_vector_type(2))) float v2f;
typedef __attribute__((ext_vector_type(8))) float v8f;

#define MATN 512
#define MATNN (MATN * MATN)
#define NBATCH 128

// ---------------------------------------------------------------------------
// Kernel 1: s[b] = ||pred_b - target_b||_F^2  (= trace(A^T A), the scale)
// float4 loads -> global_load_b128.
// ---------------------------------------------------------------------------
__global__ void frob_kernel(const float* __restrict__ pred,
                            const float* __restrict__ target,
                            float* __restrict__ s) {
  __shared__ float red[256];
  const int b = blockIdx.x;
  const int t = threadIdx.x;
  const float4* P = (const float4*)(pred + (size_t)b * MATNN);
  const float4* T = (const float4*)(target + (size_t)b * MATNN);
  float acc = 0.0f;
  for (int i = t; i < MATNN / 4; i += 256) {
    float4 p = P[i], q = T[i];
    float dx = p.x - q.x, dy = p.y - q.y, dz = p.z - q.z, dw = p.w - q.w;
    acc = fmaf(dx, dx, acc);
    acc = fmaf(dy, dy, acc);
    acc = fmaf(dz, dz, acc);
    acc = fmaf(dw, dw, acc);
  }
  red[t] = acc;
  __syncthreads();
  for (int off = 128; off > 0; off >>= 1) {
    if (t < off) red[t] += red[t + off];
    __syncthreads();
  }
  if (t == 0) s[b] = red[0];
}

// ---------------------------------------------------------------------------
// Kernel 2: G_b = (A_b^T A_b) / s[b], A = pred - target computed on the fly.
// One wave per 64x64 output tile: 4x4 register-blocked grid of 16x16 WMMA
// tiles (128 accumulator VGPRs). Per K-step of 4: 16 operand loads feed 16
// independent v_wmma_f32_16x16x4_f32 ops (no D->A/B hazards), 4x the
// arithmetic intensity of a single-tile wave.
// A-operand lane map (16x4 f32): lane L -> M = L%16, VGPR v -> K = 2*(L/16)+v
// B-operand (4x16) mirrors it; C/D: VGPR r -> M = r + 8*(L/16), N = L%16.
// ---------------------------------------------------------------------------
__global__ void ata_wmma_kernel(const float* __restrict__ pred,
                                const float* __restrict__ target,
                                const float* __restrict__ s,
                                float* __restrict__ G) {
  const int b = blockIdx.z;
  const int i0 = blockIdx.x * 64;
  const int j0 = blockIdx.y * 64;
  const int lane = threadIdx.x;
  const int g = lane >> 4;     // half-wave group
  const int m = lane & 15;     // M for A-tiles, N for B-tiles
  const float* P = pred + (size_t)b * MATNN;
  const float* T = target + (size_t)b * MATNN;

  v8f acc[4][4] = {};
  for (int k0 = 0; k0 < MATN; k0 += 4) {
    const int ra = (k0 + 2 * g) * MATN;      // row k0+2g
    const int rb = ra + MATN;                // row k0+2g+1
    v2f a[4], bm[4];
#pragma unroll
    for (int t = 0; t < 4; t++) {
      const int ci = i0 + 16 * t + m;
      const int cj = j0 + 16 * t + m;
      a[t].x  = P[ra + ci] - T[ra + ci];
      a[t].y  = P[rb + ci] - T[rb + ci];
      bm[t].x = P[ra + cj] - T[ra + cj];
      bm[t].y = P[rb + cj] - T[rb + cj];
    }
#pragma unroll
    for (int ti = 0; ti < 4; ti++)
#pragma unroll
      for (int tj = 0; tj < 4; tj++)
        acc[ti][tj] = __builtin_amdgcn_wmma_f32_16x16x4_f32(
            false, a[ti], false, bm[tj], (short)0, acc[ti][tj], false, false);
  }

  const float rs = 1.0f / s[b];
  float* Gb = G + (size_t)b * MATNN;
#pragma unroll
  for (int ti = 0; ti < 4; ti++)
#pragma unroll
    for (int tj = 0; tj < 4; tj++)
#pragma unroll
      for (int r = 0; r < 8; r++)
        Gb[(i0 + 16 * ti + r + 8 * g) * MATN + j0 + 16 * tj + m] =
            acc[ti][tj][r] * rs;
}

// ---------------------------------------------------------------------------
// Kernel 3: H_b = G_b * G_b (G symmetric => A-operand loaded via G^T rows so
// both operand streams are coalesced). Same 64x64 register blocking.
// ---------------------------------------------------------------------------
__global__ void square_wmma_kernel(const float* __restrict__ G,
                                   float* __restrict__ H) {
  const int b = blockIdx.z;
  const int i0 = blockIdx.x * 64;
  const int j0 = blockIdx.y * 64;
  const int lane = threadIdx.x;
  const int g = lane >> 4;
  const int m = lane & 15;
  const float* Gb = G + (size_t)b * MATNN;

  v8f acc[4][4] = {};
  for (int k0 = 0; k0 < MATN; k0 += 4) {
    const int ra = (k0 + 2 * g) * MATN;
    const int rb = ra + MATN;
    v2f a[4], bm[4];
#pragma unroll
    for (int t = 0; t < 4; t++) {
      const int ci = i0 + 16 * t + m;
      const int cj = j0 + 16 * t + m;
      // A_tile[M][K] = G[i0+M][k0+K] = G[k0+K][i0+M]  (symmetry)
      a[t].x  = Gb[ra + ci];
      a[t].y  = Gb[rb + ci];
      bm[t].x = Gb[ra + cj];
      bm[t].y = Gb[rb + cj];
    }
#pragma unroll
    for (int ti = 0; ti < 4; ti++)
#pragma unroll
      for (int tj = 0; tj < 4; tj++)
        acc[ti][tj] = __builtin_amdgcn_wmma_f32_16x16x4_f32(
            false, a[ti], false, bm[tj], (short)0, acc[ti][tj], false, false);
  }

  float* Hb = H + (size_t)b * MATNN;
#pragma unroll
  for (int ti = 0; ti < 4; ti++)
#pragma unroll
    for (int tj = 0; tj < 4; tj++)
#pragma unroll
      for (int r = 0; r < 8; r++)
        Hb[(i0 + 16 * ti + r + 8 * g) * MATN + j0 + 16 * tj + m] =
            acc[ti][tj][r];
}

// ---------------------------------------------------------------------------
// Kernel 4: power iteration on H_b (one 512-thread block per batch, v in LDS).
// lambda_max(H) = (sigma^2 / s)^2  =>  sigma = sqrt(s) * lambda^(1/4).
// float4 row loads -> global_load_b128; float4 LDS reads -> ds_load_b128.
// ---------------------------------------------------------------------------
__global__ void power_kernel(const float* __restrict__ H,
                             const float* __restrict__ s,
                             float* __restrict__ r) {
  __shared__ float v[MATN];
  __shared__ float red[MATN];
  const int b = blockIdx.x;
  const int t = threadIdx.x;
  const float4* row =
      (const float4*)(H + (size_t)b * MATNN + (size_t)t * MATN);
  const float4* v4 = (const float4*)v;

  v[t] = 1.0f;
  __syncthreads();

  float lam = 0.0f;
  for (int iter = 0; iter < 64; iter++) {
    float w = 0.0f;
    for (int j = 0; j < MATN / 4; j++) {
      float4 h = row[j];
      float4 x = v4[j];
      w = fmaf(h.x, x.x, w);
      w = fmaf(h.y, x.y, w);
      w = fmaf(h.z, x.z, w);
      w = fmaf(h.w, x.w, w);
    }
    red[t] = w * w;
    __syncthreads();
    for (int off = MATN / 2; off > 0; off >>= 1) {
      if (t < off) red[t] += red[t + off];
      __syncthreads();
    }
    float nrm = sqrtf(red[0]);
    lam = nrm;                    // ||H v|| with ||v||=1 (after iter 0)
    v[t] = w / nrm;
    __syncthreads();              // protects red[0] reads before next overwrite
  }

  if (t == 0) r[b] = sqrtf(s[b]) * sqrtf(sqrtf(lam));
}

// ---------------------------------------------------------------------------
// Kernel 5: out[0] = max_b r[b]
// ---------------------------------------------------------------------------
__global__ void max_kernel(const float* __restrict__ r, float* __restrict__ out) {
  __shared__ float red[NBATCH];
  const int t = threadIdx.x;
  red[t] = r[t];
  __syncthreads();
  for (int off = NBATCH / 2; off > 0; off >>= 1) {
    if (t < off) red[t] = fmaxf(red[t], red[t + off]);
    __syncthreads();
  }
  if (t == 0) out[0] = red[0];
}

extern "C" void kernel_launch(void* const* d_in, const int* in_sizes, int n_in,
                              void* d_out, int out_size, void* d_ws, size_t ws_size,
                              hipStream_t stream) {
  const float* pred = (const float*)d_in[0];
  const float* target = (const float*)d_in[1];
  float* out = (float*)d_out;

  // Workspace layout
  char* ws = (char*)d_ws;
  float* s = (float*)ws;                                   // 128 floats
  float* r = (float*)(ws + 512);                           // 128 floats
  float* G = (float*)(ws + 4096);                          // 128 MB
  float* H = (float*)(ws + 4096 + (size_t)NBATCH * MATNN * sizeof(float)); // 128 MB

  // 1. per-batch Frobenius norm^2 (scale)
  frob_kernel<<<NBATCH, 256, 0, stream>>>(pred, target, s);

  // 2. G = (A^T A)/s via fp32 WMMA, 64x64 register-blocked tiles
  dim3 gemm_grid(MATN / 64, MATN / 64, NBATCH);
  ata_wmma_kernel<<<gemm_grid, 32, 0, stream>>>(pred, target, s, G);

  // 3. H = G*G via fp32 WMMA
  square_wmma_kernel<<<gemm_grid, 32, 0, stream>>>(G, H);

  // 4. power iteration per batch -> sigma_b
  power_kernel<<<NBATCH, MATN, 0, stream>>>(H, s, r);

  // 5. global max
  max_kernel<<<1, NBATCH, 0, stream>>>(r, out);
}